// GraphConv_506806141238
// MI455X (gfx1250) — compile-verified
//
#include <hip/hip_runtime.h>

// Problem constants (match reference)
#define N_NODES 8192
#define D_INF   256
#define D_OUTF  64

// Tiling
#define SPLITK  8                    // K-split across blockIdx.y (deterministic partials)
#define KSEG    (N_NODES / SPLITK)   // 1024 K per segment
#define ROWS    128                  // output rows per workgroup (8 waves x 16 rows)
#define KC      64                   // K-chunk staged in LDS
#define MSTR    68                   // LDS stride for combined-adjacency tile (conflict-free A b64 reads)
#define SGRP    160                  // floats per K-pair group of the support tile
                                     // (pair-interleaved: (k,n) -> (k>>1)*SGRP + n*2 + (k&1);
                                     //  160 mod 64 = 32 -> half-waves hit disjoint bank halves)

typedef __attribute__((ext_vector_type(2))) float v2f;
typedef __attribute__((ext_vector_type(8))) float v8f;

// ---------------------------------------------------------------------------
// Kernel 1: support = inputs @ w + b   [8192 x 64], ~0.27 GFLOP (negligible)
// ---------------------------------------------------------------------------
__global__ __launch_bounds__(256) void support_kernel(const float* __restrict__ x,
                                                      const float* __restrict__ w,
                                                      const float* __restrict__ b,
                                                      float* __restrict__ support) {
    const int gid = blockIdx.x * 256 + threadIdx.x;
    const int m = gid >> 6;
    const int d = gid & 63;
    const float* xr = x + (size_t)m * D_INF;
    float acc = b[d];
#pragma unroll 8
    for (int k = 0; k < D_INF; ++k)
        acc = fmaf(xr[k], w[k * D_OUTF + d], acc);
    support[(size_t)m * D_OUTF + d] = acc;
}

// ---------------------------------------------------------------------------
// Kernel 2: partial[s] = (sum_a att[a]*adj_a)[rows, Kseg] @ support[Kseg, 64]
// f32 WMMA 16x16x4; combined-adjacency tile built on the fly in LDS;
// support tile stored K-pair-interleaved so every B fragment is one ds_load_b64.
// ---------------------------------------------------------------------------
__global__ __launch_bounds__(256) void spmm_wmma_kernel(const float* __restrict__ adj,
                                                        const float* __restrict__ att,
                                                        const float* __restrict__ support,
                                                        float* __restrict__ partials) {
    __shared__ float sM[ROWS * MSTR];          // combined adjacency tile [ROWS][KC] (padded)
    __shared__ float sS[(KC / 2) * SGRP];      // support tile, K-pair interleaved (20 KB)

    const int tid  = threadIdx.x;
    const int lane = tid & 31;
    const int wave = tid >> 5;

    const int rowBase = blockIdx.x * ROWS;
    const int kBase   = blockIdx.y * KSEG;

    const float a0 = att[0], a1 = att[1], a2 = att[2];

    v8f acc0 = {}, acc1 = {}, acc2 = {}, acc3 = {};

    // ISA 7.12.2 layouts (wave32):
    //  A 16x4 f32 : lane L holds A[M=L&15][K = 2*(L>>4) + {0,1}] in 2 VGPRs
    //  B 4x16 f32 : lane L holds B[K = 2*(L>>4) + {0,1}][N=L&15]
    //  C/D 16x16  : VGPR j, lane L -> M = j + 8*(L>>4), N = L&15
    const int mrow  = wave * 16 + (lane & 15);
    const int kHalf = (lane >> 4) * 2;          // 0 or 2
    const int ncol  = lane & 15;

    const float* sMa = &sM[mrow * MSTR + kHalf];          // A-fragment base (this lane)
    const float* sSb = &sS[(kHalf >> 1) * SGRP + ncol * 2]; // B-fragment base (this lane)

    for (int kc = 0; kc < KSEG; kc += KC) {
        const int k0 = kBase + kc;
        __syncthreads();   // LDS reuse guard

        // ---- stage combined adjacency tile: M = att0*A0 + att1*A1 + att2*A2
        // ROWS*KC = 8192 f32; 256 threads x float4 -> 8 passes, fully coalesced
#pragma unroll
        for (int it = 0; it < (ROWS * KC) / 1024; ++it) {
            const int i = tid * 4 + it * 1024;
            const int r = i >> 6;      // / KC
            const int c = i & 63;      // % KC (multiple of 4)
            const float* g = adj + (size_t)(rowBase + r) * N_NODES + (size_t)(k0 + c);
            const float4 x0 = *(const float4*)(g);
            const float4 x1 = *(const float4*)(g + (size_t)N_NODES * N_NODES);
            const float4 x2 = *(const float4*)(g + 2ull * N_NODES * N_NODES);
            float4 mc;
            mc.x = fmaf(a2, x2.x, fmaf(a1, x1.x, a0 * x0.x));
            mc.y = fmaf(a2, x2.y, fmaf(a1, x1.y, a0 * x0.y));
            mc.z = fmaf(a2, x2.z, fmaf(a1, x1.z, a0 * x0.z));
            mc.w = fmaf(a2, x2.w, fmaf(a1, x1.w, a0 * x0.w));
            *(float4*)&sM[r * MSTR + c] = mc;
        }
        // ---- stage support tile [KC][64], K-pair interleaved
#pragma unroll
        for (int it = 0; it < (KC * D_OUTF) / 1024; ++it) {
            const int i = tid * 4 + it * 1024;
            const int r = i >> 6;      // k within chunk
            const int c = i & 63;      // n (multiple of 4)
            const float4 s = *(const float4*)(support + (size_t)(k0 + r) * D_OUTF + c);
            float* dst = &sS[(r >> 1) * SGRP + (r & 1) + c * 2];
            dst[0] = s.x; dst[2] = s.y; dst[4] = s.z; dst[6] = s.w;
        }
        __syncthreads();

        // ---- WMMA over the chunk: 16 K-steps x 4 N-tiles per wave
        //      per step: 1 b64 (A) + 4 b64 (B) + 4 WMMA, all fragments load pre-packed
#pragma unroll
        for (int kk = 0; kk < KC; kk += 4) {
            const v2f af = *(const v2f*)(sMa + kk);              // A[mrow][kk+kHalf .. +1]
            const float* bb = sSb + (kk >> 1) * SGRP;            // group of (kk+kHalf, +1)
            const v2f b0 = *(const v2f*)(bb);
            const v2f b1 = *(const v2f*)(bb + 32);
            const v2f b2 = *(const v2f*)(bb + 64);
            const v2f b3 = *(const v2f*)(bb + 96);
            acc0 = __builtin_amdgcn_wmma_f32_16x16x4_f32(false, af, false, b0, (short)0, acc0, false, false);
            acc1 = __builtin_amdgcn_wmma_f32_16x16x4_f32(false, af, false, b1, (short)0, acc1, false, false);
            acc2 = __builtin_amdgcn_wmma_f32_16x16x4_f32(false, af, false, b2, (short)0, acc2, false, false);
            acc3 = __builtin_amdgcn_wmma_f32_16x16x4_f32(false, af, false, b3, (short)0, acc3, false, false);
        }
    }

    // ---- write this K-segment's partial result
    float* part = partials + (size_t)blockIdx.y * ((size_t)N_NODES * D_OUTF);
    const int mTop = rowBase + wave * 16 + (lane >> 4) * 8;
#pragma unroll
    for (int j = 0; j < 8; ++j) {
        float* prow = part + (size_t)(mTop + j) * D_OUTF + ncol;
        prow[0]  = acc0[j];
        prow[16] = acc1[j];
        prow[32] = acc2[j];
        prow[48] = acc3[j];
    }
}

// ---------------------------------------------------------------------------
// Kernel 3: out = sum over SPLITK partials (deterministic reduction)
// ---------------------------------------------------------------------------
__global__ __launch_bounds__(256) void reduce_kernel(const float* __restrict__ partials,
                                                     float* __restrict__ out) {
    const size_t i = ((size_t)blockIdx.x * 256 + threadIdx.x) * 4;
    const size_t stride = (size_t)N_NODES * D_OUTF;
    float4 s = *(const float4*)(partials + i);
#pragma unroll
    for (int p = 1; p < SPLITK; ++p) {
        const float4 t = *(const float4*)(partials + (size_t)p * stride + i);
        s.x += t.x; s.y += t.y; s.z += t.z; s.w += t.w;
    }
    *(float4*)((float*)out + i) = s;
}

// ---------------------------------------------------------------------------
extern "C" void kernel_launch(void* const* d_in, const int* in_sizes, int n_in,
                              void* d_out, int out_size, void* d_ws, size_t ws_size,
                              hipStream_t stream) {
    const float* inputs = (const float*)d_in[0];   // [8192, 256]
    const float* adj    = (const float*)d_in[1];   // [3, 8192, 8192]
    const float* att    = (const float*)d_in[2];   // [3]
    const float* w      = (const float*)d_in[3];   // [256, 64]
    const float* bias   = (const float*)d_in[4];   // [64]
    float* out = (float*)d_out;                    // [8192, 64]

    float* support  = (float*)d_ws;                               // 2 MB
    float* partials = support + (size_t)N_NODES * D_OUTF;         // 8 x 2 MB

    support_kernel<<<(N_NODES * D_OUTF) / 256, 256, 0, stream>>>(inputs, w, bias, support);

    dim3 grid2(N_NODES / ROWS, SPLITK);   // 64 x 8 = 512 workgroups
    spmm_wmma_kernel<<<grid2, 256, 0, stream>>>(adj, att, support, partials);

    reduce_kernel<<<(N_NODES * D_OUTF) / 1024, 256, 0, stream>>>(partials, out);
}